// PermuteMultiEmbedding_68582037782900
// MI455X (gfx1250) — compile-verified
//
#include <hip/hip_runtime.h>

// permute_multi_embedding: pure bandwidth kernel.
// Inputs  (f32, row-major): v0[B,512], v1[B,768], v2[B,256]
// Outputs (f32, concatenated flat in d_out): out0[B,640] then out1[B,896]
//   out0 row = v0[:,0:128]   | v1[:,0:256]   | v2[:,0:256]
//   out1 row = v0[:,128:384] | v1[:,256:768] | v0[:,384:512]
// All offsets/lengths are multiples of 128 floats -> every lane access is a
// 16B-aligned float4: use b128 loads/stores with TH_NT (read-once/write-once
// stream, no reuse -> don't pollute L2).

typedef __attribute__((ext_vector_type(4))) float v4f;

// Row strides in float4 units
#define V0_STR4   128   // 512 f32
#define V1_STR4   192   // 768 f32
#define V2_STR4    64   // 256 f32
#define OUT0_STR4 160   // 640 f32
#define OUT1_STR4 224   // 896 f32
#define COLS4     (OUT0_STR4 + OUT1_STR4)  // 384 f4 columns of work per row
#define ROWS_PER_BLOCK 8

__global__ __launch_bounds__(128) void
PermuteMultiEmbedding_kernel(const float* __restrict__ v0,
                             const float* __restrict__ v1,
                             const float* __restrict__ v2,
                             float* __restrict__ out,
                             int rows) {
  const int c = blockIdx.x * 128 + threadIdx.x;   // 0..383, f4 column id
  if (c >= COLS4) return;

  // ---- decode the column mapping once (compile-time-constant compares) ----
  const v4f* __restrict__ src4;
  int src_str4, src_off4;     // per-row stride / column offset in src (f4)
  size_t dst_base4;           // start of this output region in d_out (f4)
  int dst_str4, dst_off4;

  if (c < OUT0_STR4) {                 // ---- out0 region ----
    dst_base4 = 0; dst_str4 = OUT0_STR4; dst_off4 = c;
    if (c < 32)        { src4 = (const v4f*)v0; src_str4 = V0_STR4; src_off4 = c;        }
    else if (c < 96)   { src4 = (const v4f*)v1; src_str4 = V1_STR4; src_off4 = c - 32;   }
    else               { src4 = (const v4f*)v2; src_str4 = V2_STR4; src_off4 = c - 96;   }
  } else {                             // ---- out1 region ----
    const int c2 = c - OUT0_STR4;      // 0..223
    dst_base4 = (size_t)rows * OUT0_STR4; dst_str4 = OUT1_STR4; dst_off4 = c2;
    if (c2 < 64)       { src4 = (const v4f*)v0; src_str4 = V0_STR4; src_off4 = 32 + c2;  }
    else if (c2 < 192) { src4 = (const v4f*)v1; src_str4 = V1_STR4; src_off4 = c2;       } // 64 + (c2-64)
    else               { src4 = (const v4f*)v0; src_str4 = V0_STR4; src_off4 = c2 - 96;  } // 96 + (c2-192)
  }

  v4f* __restrict__ dst4 = (v4f*)out + dst_base4;

  const int row0 = blockIdx.y * ROWS_PER_BLOCK;

  // Prefetch the NEXT block's stripe toward L2 while our loads are in flight
  // (lands as global_prefetch_b8; shared L2, so cross-block prefetch is fine).
  {
    const int pr = row0 + ROWS_PER_BLOCK;
    if (pr < rows) {
      __builtin_prefetch(src4 + (size_t)pr * src_str4 + src_off4, 0, 1);
    }
  }

  // 8 independent b128 NT loads -> 8 b128 NT stores; unrolled for MLP/ILP.
#pragma unroll
  for (int r = 0; r < ROWS_PER_BLOCK; ++r) {
    const int row = row0 + r;
    if (row < rows) {
      v4f val = __builtin_nontemporal_load(src4 + (size_t)row * src_str4 + src_off4);
      __builtin_nontemporal_store(val, dst4 + (size_t)row * dst_str4 + dst_off4);
    }
  }
}

extern "C" void kernel_launch(void* const* d_in, const int* in_sizes, int n_in,
                              void* d_out, int out_size, void* d_ws, size_t ws_size,
                              hipStream_t stream) {
  const float* v0 = (const float*)d_in[0];   // [B,512]
  const float* v1 = (const float*)d_in[1];   // [B,768]
  const float* v2 = (const float*)d_in[2];   // [B,256]
  float* out = (float*)d_out;                // [B,640] ++ [B,896] flat

  const int rows = in_sizes[0] / 512;        // B (16384 in the harness)

  dim3 block(128, 1, 1);                                     // 4 wave32s
  dim3 grid(COLS4 / 128, (rows + ROWS_PER_BLOCK - 1) / ROWS_PER_BLOCK, 1);
  hipLaunchKernelGGL(PermuteMultiEmbedding_kernel, grid, block, 0, stream,
                     v0, v1, v2, out, rows);
}